// SpecAugment_89137751261497
// MI455X (gfx1250) — compile-verified
//
#include <hip/hip_runtime.h>
#include <hip/hip_bf16.h>
#include <math.h>

#define IMG_H 256
#define IMG_W 1600
#define NCTRL 128
#define NP    (NCTRL + 3)   // 131 unknowns
#define NCOLS (NP + 2)      // + 2 RHS columns = 133
#define MSTRIDE 136         // padded LDS row stride (dodge bank conflicts)

typedef __attribute__((ext_vector_type(2))) float v2f;
typedef __attribute__((ext_vector_type(8))) float v8f;

// 0.5 * ln(2): folds the phi 0.5x scale and the log2->ln conversion together.
#define HALF_LN2 0.34657359027997264f

// ---------------------------------------------------------------------------
// Kernel 1: build + solve the 131x131 TPS system (2 RHS) entirely in LDS.
// Gauss-Jordan with partial pivoting; one workgroup, ~71KB LDS (CDNA5: 320KB).
// Output: wv[0..255] = w (128x2 row-major), wv[256..261] = v (3x2 row-major).
// Runs once on one WGP -> not performance relevant; keep it simple & robust.
// ---------------------------------------------------------------------------
__global__ __launch_bounds__(256) void tps_solve_kernel(
    const float* __restrict__ src, const float* __restrict__ dst,
    float* __restrict__ wv) {
  __shared__ float M[NP][MSTRIDE];
  __shared__ int s_piv;
  const int tid = threadIdx.x;
  const int nt  = blockDim.x;

  // ---- assemble [[A, B | f], [B^T, 0 | 0]] ----
  for (int idx = tid; idx < NP * NCOLS; idx += nt) {
    const int i = idx / NCOLS, j = idx % NCOLS;
    float val;
    if (j < NCTRL) {
      const float cjy = dst[2 * j], cjx = dst[2 * j + 1];
      if (i < NCTRL) {
        const float dy = dst[2 * i] - cjy, dx = dst[2 * i + 1] - cjx;
        const float r = dy * dy + dx * dx;
        val = HALF_LN2 * r * __builtin_amdgcn_logf(fmaxf(r, 1e-10f)); // phi, order 2
      } else if (i == NCTRL)     val = cjy;              // B^T rows
      else if (i == NCTRL + 1)   val = cjx;
      else                       val = 1.0f;
    } else if (j < NP) {                                  // B columns
      if (i < NCTRL) {
        if (j == NCTRL)          val = dst[2 * i];
        else if (j == NCTRL + 1) val = dst[2 * i + 1];
        else                     val = 1.0f;
      } else                     val = 0.0f;              // zero block
    } else {                                              // RHS: flows = dst - src
      const int k = j - NP;                               // 0 = y, 1 = x
      val = (i < NCTRL) ? (dst[2 * i + k] - src[2 * i + k]) : 0.0f;
    }
    M[i][j] = val;
  }
  __syncthreads();

  // ---- Gauss-Jordan with partial pivoting ----
  for (int k = 0; k < NP; ++k) {
    if (tid == 0) {
      int p = k; float best = fabsf(M[k][k]);
      for (int i = k + 1; i < NP; ++i) {
        const float a = fabsf(M[i][k]);
        if (a > best) { best = a; p = i; }
      }
      s_piv = p;
    }
    __syncthreads();
    const int p = s_piv;
    if (p != k)
      for (int j = tid; j < NCOLS; j += nt) {
        const float t = M[k][j]; M[k][j] = M[p][j]; M[p][j] = t;
      }
    __syncthreads();
    const float piv = M[k][k];
    __syncthreads();                       // everyone reads piv before writes
    const float inv = 1.0f / piv;
    for (int j = tid; j < NCOLS; j += nt) M[k][j] *= inv;
    __syncthreads();
    for (int i = tid; i < NP; i += nt) {
      if (i == k) continue;
      const float f = M[i][k];
      for (int j = k; j < NCOLS; ++j) M[i][j] -= f * M[k][j];
    }
    __syncthreads();
  }

  // ---- emit solution: rows 0..127 -> w, rows 128..130 -> v ----
  for (int i = tid; i < NP; i += nt) {
    wv[2 * i]     = M[i][NP];
    wv[2 * i + 1] = M[i][NP + 1];
  }
}

// ---------------------------------------------------------------------------
// Kernel 2: fused TPS-apply + bilinear warp.
// Per wave32: 16 pixels. d2(control, query) tiles via V_WMMA_F32_16X16X4_F32
//   A (16x4, controls):  lanes 0-15 -> {K0=cy, K1=cx}; lanes 16-31 -> {K2=|c|^2, K3=1}
//   B (4x16, queries):   VGPR0 = rows K0|K2, VGPR1 = rows K1|K3 per half-wave
//     col n: [-2*qy, -2*qx, 1, |q|^2]  =>  A.B = |c - q|^2
//   D (16x16 f32): lane l, VGPR r -> control (r + 8*(l>=16)), query (l%16)
// phi uses raw v_log_f32 (inputs clamped >= 1e-10, never denormal) so each
// phi is 1 trans + 2 muls; the trans pipe co-executes with the WMMA pipe.
// ---------------------------------------------------------------------------
__global__ __launch_bounds__(256) void tps_warp_kernel(
    const float* __restrict__ img, const float* __restrict__ dst,
    const float* __restrict__ wv, float* __restrict__ out) {
  __shared__ float s_c[2 * NCTRL];
  __shared__ float s_w[2 * NCTRL];
  __shared__ float s_v[6];
  const int tid = threadIdx.x;
  if (tid < 2 * NCTRL) { s_c[tid] = dst[tid]; s_w[tid] = wv[tid]; }
  if (tid < 6) s_v[tid] = wv[2 * NCTRL + tid];
  __syncthreads();

  const int lane   = tid & 31;
  const int lane16 = lane & 15;
  const bool hi    = (lane >= 16);
  const int waveId = tid >> 5;
  const int m      = ((blockIdx.x * (blockDim.x >> 5) + waveId) << 4) + lane16;
  const float gy   = (float)(m / IMG_W);
  const float gx   = (float)(m % IMG_W);

  v2f b;
  b.x = hi ? 1.0f : (-2.0f * gy);                 // K0 | K2
  b.y = hi ? (gy * gy + gx * gx) : (-2.0f * gx);  // K1 | K3

  float sumy = 0.0f, sumx = 0.0f;
#pragma unroll
  for (int t = 0; t < 8; ++t) {                   // 8 tiles cover 128 controls
    const int nc   = lane16 + (t << 4);
    const float cy = s_c[2 * nc], cx = s_c[2 * nc + 1];
    v2f a;
    a.x = hi ? (cy * cy + cx * cx) : cy;          // K0 | K2
    a.y = hi ? 1.0f : cx;                         // K1 | K3
    v8f d = {0.f, 0.f, 0.f, 0.f, 0.f, 0.f, 0.f, 0.f};
    d = __builtin_amdgcn_wmma_f32_16x16x4_f32(
        /*neg_a=*/false, a, /*neg_b=*/false, b,
        /*c_mod=*/(short)0, d, /*reuse_a=*/false, /*reuse_b=*/false);
#pragma unroll
    for (int r = 0; r < 8; ++r) {
      const float d2 = d[r];
      // phi = 0.5 * d2 * ln(max(d2,eps)) = HALF_LN2 * d2 * log2(max(d2,eps))
      const float ph = HALF_LN2 * d2 * __builtin_amdgcn_logf(fmaxf(d2, 1e-10f));
      const int ci   = (t << 4) + (hi ? 8 : 0) + r;
      sumy = fmaf(ph, s_w[2 * ci],     sumy);
      sumx = fmaf(ph, s_w[2 * ci + 1], sumx);
    }
  }
  // fold the two half-wave control partitions onto lanes 0-15
  sumy += __shfl_xor(sumy, 16, 32);
  sumx += __shfl_xor(sumx, 16, 32);

  if (!hi) {
    const float fly = sumy + s_v[0] * gy + s_v[2] * gx + s_v[4];
    const float flx = sumx + s_v[1] * gy + s_v[3] * gx + s_v[5];
    const float qy = gy - fly, qx = gx - flx;
    const float fy = fminf(fmaxf(floorf(qy), 0.0f), (float)(IMG_H - 2));
    const float fx = fminf(fmaxf(floorf(qx), 0.0f), (float)(IMG_W - 2));
    const float ay = fminf(fmaxf(qy - fy, 0.0f), 1.0f);
    const float ax = fminf(fmaxf(qx - fx, 0.0f), 1.0f);
    const int iy = (int)fy, ix = (int)fx;
    const float* p = img + iy * IMG_W + ix;
    const float tl = p[0], tr = p[1], bl = p[IMG_W], br = p[IMG_W + 1];
    const float top = fmaf(ax, tr - tl, tl);
    const float bot = fmaf(ax, br - bl, bl);
    out[m] = fmaf(ay, bot - top, top);
  }
}

// ---------------------------------------------------------------------------
extern "C" void kernel_launch(void* const* d_in, const int* in_sizes, int n_in,
                              void* d_out, int out_size, void* d_ws, size_t ws_size,
                              hipStream_t stream) {
  const float* img = (const float*)d_in[0];  // (1, 256, 1600) f32
  const float* src = (const float*)d_in[1];  // (1, 128, 2) f32
  const float* dst = (const float*)d_in[2];  // (1, 128, 2) f32
  float* out = (float*)d_out;                // (1, 256, 1600, 1) f32
  float* wv  = (float*)d_ws;                 // 262 floats of scratch

  tps_solve_kernel<<<1, 256, 0, stream>>>(src, dst, wv);

  const int nPix   = IMG_H * IMG_W;          // 409600, divisible by 16
  const int nWaves = nPix / 16;              // 25600
  const int blocks = nWaves / 8;             // 3200 blocks x 8 waves
  tps_warp_kernel<<<blocks, 256, 0, stream>>>(img, dst, wv, out);
}